// VQCodebook_27668179321218
// MI455X (gfx1250) — compile-verified
//
#include <hip/hip_runtime.h>
#include <hip/hip_bf16.h>
#include <stdint.h>

typedef __attribute__((ext_vector_type(2))) float v2f;
typedef __attribute__((ext_vector_type(8))) float v8f;

#define NUM_CODES 8192
#define CODE_DIM  64
#define N_ROWS    32768   // 16 * 2048

// ---------------- Kernel 1: per-code squared norms, zero usage + SSE -------
__global__ void vq_init_kernel(const float* __restrict__ cb,
                               float* __restrict__ c_sq,
                               float* __restrict__ usage,
                               float* __restrict__ sse) {
  int i = blockIdx.x * blockDim.x + threadIdx.x;
  if (i == 0) sse[0] = 0.0f;
  if (i < NUM_CODES) {
    const float4* p = (const float4*)(cb + (size_t)i * CODE_DIM);
    float s = 0.f;
#pragma unroll
    for (int j = 0; j < CODE_DIM / 4; ++j) {
      float4 v = p[j];
      s += v.x * v.x + v.y * v.y + v.z * v.z + v.w * v.w;
    }
    c_sq[i]  = s;
    usage[i] = 0.0f;   // harness does not re-poison between replays
  }
}

// ---------------- Kernel 2: fused fp32-WMMA GEMM + argmin ------------------
// Wave owns a 16-row M-tile. A (16x64 fp32) lives in 32 VGPRs for the whole
// codebook sweep; B fragments stream from L2. Distances never hit memory.
__global__ __launch_bounds__(256)
void vq_argmin_wmma_kernel(const float* __restrict__ z,
                           const float* __restrict__ cb,
                           const float* __restrict__ c_sq,
                           int*   __restrict__ codes_i,
                           float* __restrict__ codes_f) {
  const int lane = threadIdx.x & 31;
  const int wave = threadIdx.x >> 5;
  const int half = lane >> 4;   // which 16-lane half
  const int nrow = lane & 15;   // M index for A, N index for B

  const int m0 = (blockIdx.x * 8 + wave) * 16;

  // A fragments: lane holds row m0+nrow, dims {4k + 2*half, 4k + 2*half + 1}
  const float* zrow = z + (size_t)(m0 + nrow) * CODE_DIM + 2 * half;
  v2f a[16];
#pragma unroll
  for (int k = 0; k < 16; ++k)
    a[k] = *(const v2f*)(zrow + 4 * k);

  float best[8];
  int   bidx[8];
#pragma unroll
  for (int j = 0; j < 8; ++j) { best[j] = 3.4e38f; bidx[j] = 0; }

  const float* cbl = cb + (size_t)nrow * CODE_DIM + 2 * half;

  for (int n0 = 0; n0 < NUM_CODES; n0 += 16) {
    // B fragments for this 16-code tile (one b64 load per k-step per lane)
    const float* bptr = cbl + (size_t)n0 * CODE_DIM;
    v2f b[16];
#pragma unroll
    for (int k = 0; k < 16; ++k)
      b[k] = *(const v2f*)(bptr + 4 * k);

    const float csq = c_sq[n0 + nrow];

    // D = A x B + C over the full D=64 reduction: 16 chained fp32 WMMAs
    v8f acc = {};
#pragma unroll
    for (int k = 0; k < 16; ++k)
      acc = __builtin_amdgcn_wmma_f32_16x16x4_f32(
          false, a[k], false, b[k], (short)0, acc, false, false);

    // score = ||c||^2 - 2 z.c  (||z||^2 constant per row, drops from argmin)
    const int code = n0 + nrow;
#pragma unroll
    for (int j = 0; j < 8; ++j) {
      float d = csq - 2.0f * acc[j];
      if (d < best[j]) { best[j] = d; bidx[j] = code; }  // first-hit wins ties
    }
  }

  // Reduce across the 16 lanes that share each output row (C layout:
  // VGPR j, half h -> row m0 + j + 8*h; lanes within a half span N).
#pragma unroll
  for (int j = 0; j < 8; ++j) {
#pragma unroll
    for (int s = 8; s > 0; s >>= 1) {
      float od = __shfl_xor(best[j], s, 16);
      int   oi = __shfl_xor(bidx[j], s, 16);
      if (od < best[j] || (od == best[j] && oi < bidx[j])) {
        best[j] = od; bidx[j] = oi;
      }
    }
    if (nrow == 0) {
      int row = m0 + j + 8 * half;
      codes_i[row] = bidx[j];
      codes_f[row] = (float)bidx[j];
    }
  }
}

// ---------------- Kernel 3: gather z_q, straight-through out, SSE, usage ---
__global__ __launch_bounds__(256)
void vq_gather_kernel(const float* __restrict__ z,
                      const float* __restrict__ cb,
                      const int* __restrict__ codes_i,
                      float* __restrict__ zq_out,
                      float* __restrict__ usage,
                      float* __restrict__ sse) {
  int row = blockIdx.x * blockDim.x + threadIdx.x;   // one thread per row
  int code = codes_i[row];
  const float4* zp = (const float4*)(z  + (size_t)row  * CODE_DIM);
  const float4* cp = (const float4*)(cb + (size_t)code * CODE_DIM);
  float4*       op = (float4*)(zq_out + (size_t)row * CODE_DIM);
  float s = 0.f;
#pragma unroll
  for (int j = 0; j < CODE_DIM / 4; ++j) {
    float4 zv = zp[j], cv = cp[j];
    float dx = cv.x - zv.x, dy = cv.y - zv.y;
    float dz = cv.z - zv.z, dw = cv.w - zv.w;
    float4 ov;  // z + (z_q - z): matches reference rounding
    ov.x = zv.x + dx; ov.y = zv.y + dy; ov.z = zv.z + dz; ov.w = zv.w + dw;
    op[j] = ov;
    s += dx * dx + dy * dy + dz * dz + dw * dw;
  }
  atomicAdd(usage + code, 1.0f);
  // wave-level SSE reduction, one atomic per wave
#pragma unroll
  for (int o = 16; o > 0; o >>= 1) s += __shfl_xor(s, o, 32);
  if ((threadIdx.x & 31) == 0) atomicAdd(sse, s);
}

// ---------------- Kernel 4: finalize scalar losses -------------------------
__global__ void vq_final_kernel(const float* __restrict__ sse,
                                float* __restrict__ losses) {
  float mean = sse[0] * (1.0f / (float)((size_t)N_ROWS * CODE_DIM));
  losses[0] = mean;                 // codebook_loss
  losses[1] = mean * 0.25f;         // commitment_loss
  losses[2] = mean + mean * 0.25f;  // total_vq
}

extern "C" void kernel_launch(void* const* d_in, const int* in_sizes, int n_in,
                              void* d_out, int out_size, void* d_ws, size_t ws_size,
                              hipStream_t stream) {
  const float* z  = (const float*)d_in[0];   // [16,2048,64]
  const float* cb = (const float*)d_in[1];   // [8192,64]

  float* out     = (float*)d_out;
  float* zq_st   = out;                                  // 32768*64
  float* codes_f = out + (size_t)N_ROWS * CODE_DIM;      // 32768
  float* losses  = codes_f + N_ROWS;                     // 3 scalars
  float* usage   = losses + 3;                           // 8192

  float* c_sq    = (float*)d_ws;                         // 8192 f32
  int*   codes_i = (int*)(c_sq + NUM_CODES);             // 32768 i32
  float* sse     = (float*)(codes_i + N_ROWS);           // 1 f32

  vq_init_kernel<<<NUM_CODES / 256, 256, 0, stream>>>(cb, c_sq, usage, sse);
  vq_argmin_wmma_kernel<<<N_ROWS / 128, 256, 0, stream>>>(z, cb, c_sq,
                                                          codes_i, codes_f);
  vq_gather_kernel<<<N_ROWS / 256, 256, 0, stream>>>(z, cb, codes_i,
                                                     zq_st, usage, sse);
  vq_final_kernel<<<1, 1, 0, stream>>>(sse, losses);
}